// NGCFLayer_32341103739241
// MI455X (gfx1250) — compile-verified
//
#include <hip/hip_runtime.h>
#include <math.h>

typedef __attribute__((ext_vector_type(2))) float v2f;
typedef __attribute__((ext_vector_type(8))) float v8f;

#define NUSERS 500000
#define NITEMS 200000
#define NEDGES 1500000
#define LRELU  0.2f
#define EPSN   1e-12f

// D = A(16x4 f32) * B(4x16 f32) + C(16x16 f32)
// 8-arg pattern: (neg_a, A, neg_b, B, c_mod, C, reuse_a, reuse_b)
__device__ __forceinline__ v8f wmma4_f32(v2f a, v2f b, v8f c) {
    return __builtin_amdgcn_wmma_f32_16x16x4_f32(
        false, a, false, b, (short)0, c, false, false);
}

// ---------------------------------------------------------------------------
// Edge kernel: one wave32 per tile of 16 edges.
// Fused messages as two K=128 GEMMs against stacked [W1;W2]:
//   accUP = xu@W1 + (xu*xi)@W2 + (b1+b2)   -> nui * accUP scattered to items
//   accIP = xi@W1 + (xu*xi)@W2 + (b1+b2)   -> niu * accIP scattered to users
// ---------------------------------------------------------------------------
__global__ __launch_bounds__(256) void ngcf_edge_kernel(
    const float* __restrict__ fu,  const float* __restrict__ fi,
    const float* __restrict__ nui, const float* __restrict__ niu,
    const float* __restrict__ W1,  const float* __restrict__ b1,
    const float* __restrict__ W2,  const float* __restrict__ b2,
    const int*   __restrict__ src, const int*   __restrict__ dst,
    float*       __restrict__ out)
{
    // Transposed weights: Wt[n*64 + k] = W[k][n]  (B-fragment = aligned b64 load)
    __shared__ float W1t[64 * 64];
    __shared__ float W2t[64 * 64];
    __shared__ float bsum[64];
    for (int idx = threadIdx.x; idx < 4096; idx += 256) {
        const int n = idx >> 6, k = idx & 63;
        W1t[idx] = W1[k * 64 + n];
        W2t[idx] = W2[k * 64 + n];
    }
    if (threadIdx.x < 64) bsum[threadIdx.x] = b1[threadIdx.x] + b2[threadIdx.x];
    __syncthreads();

    const int lane = threadIdx.x & 31;
    const int wv   = threadIdx.x >> 5;
    const int col  = lane & 15;   // A-row index / B-N column within 16-tile
    const int hi   = lane >> 4;   // half-wave selector
    const int waveGlobal = blockIdx.x * 8 + wv;
    const int totalWaves = gridDim.x * 8;
    const int nTiles = NEDGES / 16;   // 93750, exact

    for (int tile = waveGlobal; tile < nTiles; tile += totalWaves) {
        // prefetch next tile's edge metadata (uniform branch; EXEC stays full)
        const int nextTile = tile + totalWaves;
        if (nextTile < nTiles) {
            const int en = nextTile * 16 + col;
            __builtin_prefetch(src + en, 0, 3);
            __builtin_prefetch(dst + en, 0, 3);
            __builtin_prefetch(nui + en, 0, 3);
            __builtin_prefetch(niu + en, 0, 3);
        }

        const int   e     = tile * 16 + col;
        const int   mySrc = src[e];
        const int   myDst = dst[e];
        const float myNui = nui[e];
        const float myNiu = niu[e];
        // A fragment: lane holds A[m=col][K = 4kk + 2*hi + {0,1}]
        const float* rowU = fu + (long)mySrc * 64 + 2 * hi;
        const float* rowI = fi + (long)myDst * 64 + 2 * hi;

        v8f accUP[4], accIP[4];
        #pragma unroll
        for (int t = 0; t < 4; ++t) {
            const float bb = bsum[t * 16 + col];
            #pragma unroll
            for (int i = 0; i < 8; ++i) { accUP[t][i] = bb; accIP[t][i] = bb; }
        }

        #pragma unroll 2
        for (int kk = 0; kk < 16; ++kk) {
            const int k0 = kk * 4;
            const int kr = k0 + 2 * hi;
            v2f au; au.x = rowU[k0]; au.y = rowU[k0 + 1];
            v2f ai; ai.x = rowI[k0]; ai.y = rowI[k0 + 1];
            v2f ap; ap.x = au.x * ai.x; ap.y = au.y * ai.y;
            #pragma unroll
            for (int t = 0; t < 4; ++t) {
                const int cc = t * 16 + col;
                const v2f bf1 = *(const v2f*)(W1t + cc * 64 + kr);
                const v2f bf2 = *(const v2f*)(W2t + cc * 64 + kr);
                accUP[t] = wmma4_f32(au, bf1, accUP[t]);
                accUP[t] = wmma4_f32(ap, bf2, accUP[t]);
                accIP[t] = wmma4_f32(ai, bf1, accIP[t]);
                accIP[t] = wmma4_f32(ap, bf2, accIP[t]);
            }
        }

        // Scatter: lane holds rows m = 8*hi + i, cols t*16+col
        #pragma unroll
        for (int i = 0; i < 8; ++i) {
            const int   m   = hi * 8 + i;
            const int   sR  = __shfl(mySrc, m, 32);
            const int   dR  = __shfl(myDst, m, 32);
            const float wui = __shfl(myNui, m, 32);
            const float wiu = __shfl(myNiu, m, 32);
            float* itemRow = out + (long)(NUSERS + dR) * 64;
            float* userRow = out + (long)sR * 64;
            #pragma unroll
            for (int t = 0; t < 4; ++t) {
                const int cc = t * 16 + col;
                atomicAdd(itemRow + cc, wui * accUP[t][i]);
                atomicAdd(userRow + cc, wiu * accIP[t][i]);
            }
        }
    }
}

// ---------------------------------------------------------------------------
// Post kernel: h = segment_sum(in out) + feat @ W1 + b1 ; LeakyReLU ; L2-norm.
// One wave per 16-node tile; NU and NI are multiples of 16 (no straddling).
// ---------------------------------------------------------------------------
__global__ __launch_bounds__(256) void ngcf_post_kernel(
    const float* __restrict__ fu, const float* __restrict__ fi,
    const float* __restrict__ W1, const float* __restrict__ b1,
    float*       __restrict__ out)
{
    __shared__ float W1t[64 * 64];   // transposed: W1t[n*64 + k] = W1[k][n]
    __shared__ float b1s[64];
    for (int idx = threadIdx.x; idx < 4096; idx += 256) {
        const int n = idx >> 6, k = idx & 63;
        W1t[idx] = W1[k * 64 + n];
    }
    if (threadIdx.x < 64) b1s[threadIdx.x] = b1[threadIdx.x];
    __syncthreads();

    const int lane = threadIdx.x & 31;
    const int wv   = threadIdx.x >> 5;
    const int col  = lane & 15;
    const int hi   = lane >> 4;
    const int waveGlobal = blockIdx.x * 8 + wv;
    const int totalWaves = gridDim.x * 8;
    const int nTiles = (NUSERS + NITEMS) / 16;   // 43750, exact

    for (int tile = waveGlobal; tile < nTiles; tile += totalWaves) {
        const int g = tile * 16 + col;
        const float* row = (g < NUSERS) ? (fu + (long)g * 64 + 2 * hi)
                                        : (fi + (long)(g - NUSERS) * 64 + 2 * hi);
        v8f acc[4];
        #pragma unroll
        for (int t = 0; t < 4; ++t) {
            const float bb = b1s[t * 16 + col];
            #pragma unroll
            for (int i = 0; i < 8; ++i) acc[t][i] = bb;
        }

        #pragma unroll 4
        for (int kk = 0; kk < 16; ++kk) {
            const int k0 = kk * 4;
            const int kr = k0 + 2 * hi;
            v2f a; a.x = row[k0]; a.y = row[k0 + 1];
            #pragma unroll
            for (int t = 0; t < 4; ++t) {
                const int cc = t * 16 + col;
                const v2f bf = *(const v2f*)(W1t + cc * 64 + kr);
                acc[t] = wmma4_f32(a, bf, acc[t]);
            }
        }

        // add scattered sums, LeakyReLU, L2-normalize per row, store in place
        #pragma unroll
        for (int i = 0; i < 8; ++i) {
            const int  m  = hi * 8 + i;
            const long gr = (long)(tile * 16 + m) * 64;
            float s = 0.0f;
            #pragma unroll
            for (int t = 0; t < 4; ++t) {
                float v = acc[t][i] + out[gr + t * 16 + col];
                v = (v > 0.0f) ? v : LRELU * v;
                acc[t][i] = v;
                s += v * v;
            }
            // reduce sum-of-squares over the 16 lanes holding this row
            s += __shfl_xor(s, 1, 32);
            s += __shfl_xor(s, 2, 32);
            s += __shfl_xor(s, 4, 32);
            s += __shfl_xor(s, 8, 32);
            const float rs = 1.0f / fmaxf(sqrtf(s), EPSN);
            #pragma unroll
            for (int t = 0; t < 4; ++t)
                out[gr + t * 16 + col] = acc[t][i] * rs;
        }
    }
}

extern "C" void kernel_launch(void* const* d_in, const int* in_sizes, int n_in,
                              void* d_out, int out_size, void* d_ws, size_t ws_size,
                              hipStream_t stream) {
    const float* fu  = (const float*)d_in[0];
    const float* fi  = (const float*)d_in[1];
    const float* nui = (const float*)d_in[2];
    const float* niu = (const float*)d_in[3];
    const float* W1  = (const float*)d_in[4];
    const float* b1  = (const float*)d_in[5];
    const float* W2  = (const float*)d_in[6];
    const float* b2  = (const float*)d_in[7];
    const int*   src = (const int*)d_in[8];
    const int*   dst = (const int*)d_in[9];
    float* out = (float*)d_out;

    // d_out doubles as the segment-sum accumulator: zero it every call.
    hipMemsetAsync(out, 0, (size_t)out_size * sizeof(float), stream);

    ngcf_edge_kernel<<<2048, 256, 0, stream>>>(fu, fi, nui, niu,
                                               W1, b1, W2, b2, src, dst, out);
    ngcf_post_kernel<<<1024, 256, 0, stream>>>(fu, fi, W1, b1, out);
}